// FourierTransformLayer_88218628260403
// MI455X (gfx1250) — compile-verified
//
#include <hip/hip_runtime.h>
#include <hip/hip_bf16.h>

// ---------------------------------------------------------------------------
// FourierTransformLayer on MI455X (gfx1250):
//   out = Re( conj(F) * ((F * X * F) .* M) * conj(F) ) / N^2
// F = 512x512 symmetric DFT matrix; M = ifftshift(mask * param).
// 2D (I)DFTs are 512^3 f32 GEMMs on V_WMMA_F32_16X16X4_F32, with A/mask tiles
// staged via GLOBAL_LOAD_ASYNC_TO_LDS_B128 (double-buffered, ASYNCcnt).
// ---------------------------------------------------------------------------

#define N 512
#define NN (N * N)          // 262144 elements per plane
#define NIMG 96             // 32 batch * 3 channels
#define HALF 256
#define KBLK 32
#define APAD 36             // LDS A row stride (float4-aligned, bank-clean)

typedef __attribute__((ext_vector_type(2))) float v2f;
typedef __attribute__((ext_vector_type(8))) float v8f;
typedef __attribute__((ext_vector_type(4))) int v4i;

// Builtin wants: arg0 = v4i in addrspace(1) (global), arg1 = LDS destination.
typedef __attribute__((address_space(1))) v4i* gvp_t;
typedef __attribute__((address_space(3))) v4i* lvp_t;

static __device__ __forceinline__ v8f wmma_f32(v2f a, v2f b, v8f c) {
  // V_WMMA_F32_16X16X4_F32 : D = A(16x4) * B(4x16) + C(16x16)
  return __builtin_amdgcn_wmma_f32_16x16x4_f32(
      false, a, false, b, (short)0, c, false, false);
}

static __device__ __forceinline__ void async_copy_b128(const float* g,
                                                       float* l) {
  __builtin_amdgcn_global_load_async_to_lds_b128((gvp_t)g, (lvp_t)l, 0, 0);
}

static __device__ __forceinline__ void wait_async0() {
  asm volatile("s_wait_asynccnt 0x0" ::: "memory");
}

// ---------------------------------------------------------------------------
// Table builder: DFT twiddles + fused (mask * param) in unshifted coords.
// ---------------------------------------------------------------------------
__global__ void ftl_build_tables(float* __restrict__ Fr, float* __restrict__ Fi,
                                 float* __restrict__ Fcj,
                                 const float* __restrict__ param,
                                 float* __restrict__ M) {
  int id = blockIdx.x * blockDim.x + threadIdx.x;
  if (id >= NN) return;
  int u = id >> 9;
  int v = id & (N - 1);

  int k = (u * v) & (N - 1);  // reduce mod N first for float precision
  float theta = -6.2831853071795864769f * (float)k / (float)N;
  float s, c;
  __sincosf(theta, &s, &c);
  Fr[id] = c;
  Fi[id] = s;       // = -sin(2*pi*k/N)  (forward DFT)
  Fcj[id] = -s;     // imag of conj(F)

  int su = (u + HALF) & (N - 1);
  int sv = (v + HALF) & (N - 1);
  float dh = fabsf((float)(su - HALF));
  float dw = fabsf((float)(sv - HALF));
  bool center = (dh * dh + dw * dw) <= 1.0f;
  float nh = dh / (float)HALF;
  float nw = dw / (float)HALF;
  float dist = sqrtf(nh * nh + nw * nw);
  const float SMOOTH = 0.5f;
  float m = center ? SMOOTH : fmaxf(1.0f, dist / SMOOTH);

  int sidx = su * N + sv;
#pragma unroll
  for (int ch = 0; ch < 3; ++ch) M[ch * NN + id] = m * param[ch * NN + sidx];
}

// ---------------------------------------------------------------------------
// Complex GEMM:  C = (A [.* Mplane]) * B, 512x512 row-major, per-image batch.
// Block = 128 threads (4 wave32), WG tile 64x64, wave tile 32x32 (2x2 WMMA),
// K panels of 32 double-buffered:
//   A (+mask) tiles : GLOBAL_LOAD_ASYNC_TO_LDS_B128, layout [64][36]
//   B tiles         : global->VGPR->ds_store, k-pair interleaved [16][64][2]
//                     so each B fragment is one aligned ds_load_b64
// ---------------------------------------------------------------------------
template <bool A_CPLX, bool B_CPLX, bool WRITE_IMAG, bool PREMUL>
__global__ __launch_bounds__(128)
void ftl_cgemm(const float* __restrict__ Ar, const float* __restrict__ Ai,
               long long aStride,
               const float* __restrict__ Br, const float* __restrict__ Bi,
               long long bStride,
               const float* __restrict__ Mbase,
               float* __restrict__ Cr, float* __restrict__ Ci,
               long long cStride, float scale) {
  __shared__ float lsA[2][A_CPLX ? 2 : 1][64][APAD];
  __shared__ float lsB[2][B_CPLX ? 2 : 1][KBLK / 2][64][2];
  __shared__ float lsM[PREMUL ? 2 : 1][64][APAD];

  const int img = blockIdx.y;
  const float* ar = Ar + (long long)img * aStride;
  const float* ai = A_CPLX ? (Ai + (long long)img * aStride) : nullptr;
  const float* br = Br + (long long)img * bStride;
  const float* bi = B_CPLX ? (Bi + (long long)img * bStride) : nullptr;
  const float* mp = PREMUL ? (Mbase + (long long)(img % 3) * NN) : nullptr;

  const int tr = (blockIdx.x >> 3) * 64;
  const int tc = (blockIdx.x & 7) * 64;

  const int tid = threadIdx.x;
  const int lane = tid & 31;
  const int wave = tid >> 5;
  const int wRow = (wave >> 1) * 32;
  const int wCol = (wave & 1) * 32;
  const int laneHi = lane >> 4;
  const int laneLo = lane & 15;

  v8f accR[2][2], accI[2][2];
#pragma unroll
  for (int i = 0; i < 2; ++i)
#pragma unroll
    for (int j = 0; j < 2; ++j) {
      accR[i][j] = (v8f){};
      if constexpr (WRITE_IMAG) accI[i][j] = (v8f){};
    }

  // B register staging for the next panel (2 slots/thread, 2 k-rows each)
  float4 bR0[2], bR1[2], bI0[2], bI1[2];

  // --- helpers (inlined) ---------------------------------------------------
  auto issueA = [&](int kb, int buf) {
#pragma unroll
    for (int i = 0; i < 4; ++i) {
      int id = tid + i * 128;     // 512 float4 slots over 64x32 tile
      int row = id >> 3;
      int kk = (id & 7) * 4;
      long long goff = (long long)(tr + row) * N + kb + kk;
      async_copy_b128(ar + goff, &lsA[buf][0][row][kk]);
      if constexpr (A_CPLX) async_copy_b128(ai + goff, &lsA[buf][1][row][kk]);
      if constexpr (PREMUL) async_copy_b128(mp + goff, &lsM[buf][row][kk]);
    }
  };
  auto loadB = [&](int kb) {
#pragma unroll
    for (int j = 0; j < 2; ++j) {
      int s = tid + j * 128;      // 256 slots: 16 k-pairs x 16 col-quads
      int t = s >> 4;
      int c4 = (s & 15) * 4;
      long long g = (long long)(kb + 2 * t) * N + tc + c4;
      bR0[j] = *(const float4*)&br[g];
      bR1[j] = *(const float4*)&br[g + N];
      if constexpr (B_CPLX) {
        bI0[j] = *(const float4*)&bi[g];
        bI1[j] = *(const float4*)&bi[g + N];
      }
    }
  };
  auto storeB = [&](int buf) {
#pragma unroll
    for (int j = 0; j < 2; ++j) {
      int s = tid + j * 128;
      int t = s >> 4;
      int c4 = (s & 15) * 4;
      const float* r0 = (const float*)&bR0[j];
      const float* r1 = (const float*)&bR1[j];
      const float* i0 = (const float*)&bI0[j];
      const float* i1 = (const float*)&bI1[j];
#pragma unroll
      for (int e = 0; e < 4; ++e) {
        *(float2*)&lsB[buf][0][t][c4 + e][0] = make_float2(r0[e], r1[e]);
        if constexpr (B_CPLX)
          *(float2*)&lsB[buf][1][t][c4 + e][0] = make_float2(i0[e], i1[e]);
      }
    }
  };
  // -------------------------------------------------------------------------

  // prologue: stage panel 0 into buffer 0
  issueA(0, 0);
  loadB(0);
  storeB(0);

  const int NSTAGE = N / KBLK;  // 16
  for (int stage = 0; stage < NSTAGE; ++stage) {
    const int cur = stage & 1;
    wait_async0();        // this wave's async tiles for `cur` have landed
    __syncthreads();      // everyone's async tiles + ds_stores visible

    if (stage + 1 < NSTAGE) {
      issueA((stage + 1) * KBLK, cur ^ 1);  // overlap DMA with WMMA below
      loadB((stage + 1) * KBLK);
    }

    // ---- WMMA over the 32-wide K panel, step 4 ----
#pragma unroll
    for (int kk = 0; kk < KBLK; kk += 4) {
      const int ko = kk + laneHi * 2;        // A frag: VGPR0=K0|K2, VGPR1=K1|K3
      const int t = (kk >> 1) + laneHi;      // B pair row
      v2f fAr[2], fAi[2], fAni[2];
#pragma unroll
      for (int mg = 0; mg < 2; ++mg) {
        const int mrow = wRow + mg * 16 + laneLo;
        fAr[mg] = *(const v2f*)&lsA[cur][0][mrow][ko];
        v2f m2;
        if constexpr (PREMUL) {
          m2 = *(const v2f*)&lsM[cur][mrow][ko];
          fAr[mg] *= m2;
        }
        if constexpr (A_CPLX) {
          fAi[mg] = *(const v2f*)&lsA[cur][1][mrow][ko];
          if constexpr (PREMUL) fAi[mg] *= m2;
          fAni[mg] = -fAi[mg];  // VALU negate (f32 WMMA lacks A-neg)
        }
      }
      v2f fBr[2], fBi[2];
#pragma unroll
      for (int ng = 0; ng < 2; ++ng) {
        const int ncol = wCol + ng * 16 + laneLo;
        fBr[ng] = *(const v2f*)&lsB[cur][0][t][ncol][0];  // one ds_load_b64
        if constexpr (B_CPLX)
          fBi[ng] = *(const v2f*)&lsB[cur][1][t][ncol][0];
      }
#pragma unroll
      for (int mg = 0; mg < 2; ++mg)
#pragma unroll
        for (int ng = 0; ng < 2; ++ng) {
          accR[mg][ng] = wmma_f32(fAr[mg], fBr[ng], accR[mg][ng]);
          if constexpr (A_CPLX && B_CPLX)
            accR[mg][ng] = wmma_f32(fAni[mg], fBi[ng], accR[mg][ng]);
          if constexpr (WRITE_IMAG) {
            if constexpr (B_CPLX)
              accI[mg][ng] = wmma_f32(fAr[mg], fBi[ng], accI[mg][ng]);
            if constexpr (A_CPLX)
              accI[mg][ng] = wmma_f32(fAi[mg], fBr[ng], accI[mg][ng]);
          }
        }
    }

    if (stage + 1 < NSTAGE) storeB(cur ^ 1);  // next-panel B into other buffer
  }

  // ---- epilogue: C/D layout — VGPR r: lanes0-15 M=r, lanes16-31 M=8+r ----
  float* cr = Cr + (long long)img * cStride;
  float* ci = WRITE_IMAG ? (Ci + (long long)img * cStride) : nullptr;
#pragma unroll
  for (int mg = 0; mg < 2; ++mg) {
    const int rowBase = tr + wRow + mg * 16 + laneHi * 8;
#pragma unroll
    for (int ng = 0; ng < 2; ++ng) {
      const int col = tc + wCol + ng * 16 + laneLo;
#pragma unroll
      for (int r = 0; r < 8; ++r) {
        const long long off = (long long)(rowBase + r) * N + col;
        cr[off] = scale * accR[mg][ng][r];
        if constexpr (WRITE_IMAG) ci[off] = scale * accI[mg][ng][r];
      }
    }
  }
}

// ---------------------------------------------------------------------------
// Launch: 4 GEMM stages over ping-pong workspace buffers.
// ws layout (floats): Fr | Fi | Fcj | M[3] | b0r | b0i | b1r | b1i
// ---------------------------------------------------------------------------
extern "C" void kernel_launch(void* const* d_in, const int* in_sizes, int n_in,
                              void* d_out, int out_size, void* d_ws, size_t ws_size,
                              hipStream_t stream) {
  const float* x = (const float*)d_in[0];      // [32,3,512,512]
  const float* param = (const float*)d_in[1];  // [3,512,512]
  float* out = (float*)d_out;                  // [32,3,512,512]

  float* ws = (float*)d_ws;
  const long long PB = (long long)NIMG * NN;
  float* Fr  = ws;
  float* Fi  = Fr + NN;
  float* Fcj = Fi + NN;
  float* M   = Fcj + NN;
  float* b0r = M + 3LL * NN;
  float* b0i = b0r + PB;
  float* b1r = b0i + PB;
  float* b1i = b1r + PB;

  ftl_build_tables<<<NN / 256, 256, 0, stream>>>(Fr, Fi, Fcj, param, M);

  const dim3 grid(64, NIMG);
  const dim3 blk(128);
  const float inv = 1.0f / (float)NN;

  // T1 = X * F            (A real per-image, B complex shared)
  ftl_cgemm<false, true, true, false><<<grid, blk, 0, stream>>>(
      x, nullptr, (long long)NN, Fr, Fi, 0LL, nullptr,
      b0r, b0i, (long long)NN, 1.0f);

  // Y = F * T1            (A complex shared, B complex per-image)
  ftl_cgemm<true, true, true, false><<<grid, blk, 0, stream>>>(
      Fr, Fi, 0LL, b0r, b0i, (long long)NN, nullptr,
      b1r, b1i, (long long)NN, 1.0f);

  // T2 = (Y .* M) * conj(F)
  ftl_cgemm<true, true, true, true><<<grid, blk, 0, stream>>>(
      b1r, b1i, (long long)NN, Fr, Fcj, 0LL, M,
      b0r, b0i, (long long)NN, 1.0f);

  // out = Re( conj(F) * T2 ) / N^2
  ftl_cgemm<true, true, false, false><<<grid, blk, 0, stream>>>(
      Fr, Fcj, 0LL, b0r, b0i, (long long)NN, nullptr,
      out, nullptr, (long long)NN, inv);
}